// HierAttNet_59889023975737
// MI455X (gfx1250) — compile-verified
//
#include <hip/hip_runtime.h>

typedef __attribute__((ext_vector_type(2))) float v2f;
typedef __attribute__((ext_vector_type(8))) float v8f;

#define B_    64
#define S_    30
#define W_    35
#define N_    (S_ * W_)   // 1050
#define D_    300
#define K_    2048
#define NODE_ 5000

// ---------------------------------------------------------------------------
// Kernel 1: attn_score[b,n] = word_attn[b,s,w] * sent_attn[b,s]  (also output)
//           ctx[b,d]       = sum_n attn_score[b,n] * emb[doc_ids[b,n], d]
// One block per batch element; 8 waves stride over n; lanes own d-columns.
// ---------------------------------------------------------------------------
__global__ __launch_bounds__(256) void k_attn_ctx(
    const float* __restrict__ word_attn, const float* __restrict__ sent_attn,
    const int* __restrict__ doc_ids, const float* __restrict__ emb,
    float* __restrict__ attn_out, float* __restrict__ ctx)
{
    __shared__ float s_attn[N_];
    __shared__ int   s_idx[N_];
    __shared__ float s_ctx[D_];

    const int b   = blockIdx.x;
    const int tid = threadIdx.x;

    for (int n = tid; n < N_; n += 256) {
        const int s = n / W_;
        const float a = word_attn[b * N_ + n] * sent_attn[b * S_ + s];
        s_attn[n] = a;
        attn_out[b * N_ + n] = a;
        s_idx[n] = doc_ids[b * N_ + n];
    }
    if (tid < D_) s_ctx[tid] = 0.0f;
    __syncthreads();

    const int wave = tid >> 5;
    const int lane = tid & 31;

    float acc[10];
#pragma unroll
    for (int j = 0; j < 10; ++j) acc[j] = 0.0f;

    for (int n = wave; n < N_; n += 8) {
        const float a = s_attn[n];
        const float* __restrict__ row = emb + (size_t)s_idx[n] * D_;
#pragma unroll
        for (int j = 0; j < 9; ++j)
            acc[j] = fmaf(a, row[lane + 32 * j], acc[j]);
        if (lane < D_ - 288)
            acc[9] = fmaf(a, row[lane + 288], acc[9]);
    }
#pragma unroll
    for (int j = 0; j < 9; ++j)
        atomicAdd(&s_ctx[lane + 32 * j], acc[j]);   // ds_add_f32
    if (lane < D_ - 288)
        atomicAdd(&s_ctx[lane + 288], acc[9]);
    __syncthreads();

    if (tid < D_) ctx[b * D_ + tid] = s_ctx[tid];
}

// ---------------------------------------------------------------------------
// Kernel 2: weighted[64,2048] = ctx[64,300] @ VvT[300,2048]
// One wave per 16x16 tile, fp32 WMMA 16x16x4. 512 tiles = 64 blocks x 8 waves.
// ---------------------------------------------------------------------------
__global__ __launch_bounds__(256) void k_gemm1(
    const float* __restrict__ ctx, const float* __restrict__ VvT,
    float* __restrict__ weighted)
{
    const int wave = threadIdx.x >> 5;
    const int lane = threadIdx.x & 31;
    const int tile = blockIdx.x * 8 + wave;   // [0,512)
    const int mt   = tile >> 7;               // /128 -> [0,4)
    const int nt   = tile & 127;              // [0,128)
    const int m0 = mt * 16, n0 = nt * 16;

    const int half = lane >> 4;               // K-half select
    const int l15  = lane & 15;
    const int arow = m0 + l15;                // A: M = lane&15
    const int col  = n0 + l15;                // B/D: N = lane&15

    v8f acc = {};
    for (int k0 = 0; k0 < D_; k0 += 4) {
        const int kk = k0 + 2 * half;
        v2f a = *(const v2f*)(ctx + arow * D_ + kk);   // A[M, kk..kk+1]
        v2f bf;
        bf.x = VvT[(size_t)kk       * K_ + col];       // B[kk  , N]
        bf.y = VvT[(size_t)(kk + 1) * K_ + col];       // B[kk+1, N]
        acc = __builtin_amdgcn_wmma_f32_16x16x4_f32(
            false, a, false, bf, (short)0, acc, false, false);
    }
#pragma unroll
    for (int i = 0; i < 8; ++i) {
        const int row = m0 + half * 8 + i;             // C/D layout
        weighted[row * K_ + col] = acc[i];
    }
}

// ---------------------------------------------------------------------------
// Kernel 3: final[64,5000] = weighted[64,2048] @ path_sem[5000,2048]^T
// B^T layout means each lane reads a contiguous float2 from one path_sem row;
// path_sem (40MB) is L2-resident and reused across the 4 M-tiles.
// ---------------------------------------------------------------------------
__global__ __launch_bounds__(256) void k_gemm2(
    const float* __restrict__ weighted, const float* __restrict__ path_sem,
    float* __restrict__ out)
{
    const int NT = (NODE_ + 15) / 16;         // 313 N-tiles
    const int wave = threadIdx.x >> 5;
    const int lane = threadIdx.x & 31;
    const int tile = blockIdx.x * 8 + wave;
    if (tile >= 4 * NT) return;               // wave-uniform: EXEC stays all-1s

    const int mt = tile / NT;
    const int nt = tile % NT;
    const int m0 = mt * 16, n0 = nt * 16;

    const int half = lane >> 4;
    const int l15  = lane & 15;
    const int arow = m0 + l15;
    const int col  = n0 + l15;
    const int bn   = (col < NODE_) ? col : (NODE_ - 1);  // clamp tail loads

    const float* __restrict__ aptr = weighted + (size_t)arow * K_ + 2 * half;
    const float* __restrict__ bptr = path_sem + (size_t)bn   * K_ + 2 * half;

    v8f acc = {};
    for (int k0 = 0; k0 < K_; k0 += 4) {
        v2f a = *(const v2f*)(aptr + k0);     // A[M, k..k+1]
        v2f b = *(const v2f*)(bptr + k0);     // B[k..k+1, N] = path_sem[N, k..k+1]
        acc = __builtin_amdgcn_wmma_f32_16x16x4_f32(
            false, a, false, b, (short)0, acc, false, false);
    }
    if (col < NODE_) {
#pragma unroll
        for (int i = 0; i < 8; ++i) {
            const int row = m0 + half * 8 + i;
            out[row * NODE_ + col] = acc[i];
        }
    }
}

// ---------------------------------------------------------------------------
extern "C" void kernel_launch(void* const* d_in, const int* in_sizes, int n_in,
                              void* d_out, int out_size, void* d_ws, size_t ws_size,
                              hipStream_t stream) {
    (void)in_sizes; (void)n_in; (void)out_size; (void)ws_size;

    const float* word_attn = (const float*)d_in[0];   // [64,30,35]
    const float* sent_attn = (const float*)d_in[1];   // [64,30]
    const int*   doc_ids   = (const int*)  d_in[2];   // [64,30,35]
    const float* emb_table = (const float*)d_in[3];   // [50001,300]
    const float* VvT       = (const float*)d_in[4];   // [300,2048]
    const float* path_sem  = (const float*)d_in[5];   // [5000,2048]

    float* final_score = (float*)d_out;               // [64,5000]
    float* attn_out    = final_score + B_ * NODE_;    // [64,1050]

    float* ctx      = (float*)d_ws;                   // [64,300]
    float* weighted = ctx + B_ * D_;                  // [64,2048]

    k_attn_ctx<<<B_, 256, 0, stream>>>(word_attn, sent_attn, doc_ids,
                                       emb_table, attn_out, ctx);
    k_gemm1<<<64, 256, 0, stream>>>(ctx, VvT, weighted);

    const int tiles = 4 * ((NODE_ + 15) / 16);        // 1252
    k_gemm2<<<(tiles + 7) / 8, 256, 0, stream>>>(weighted, path_sem, final_score);
}